// QuantumLayer_7146825581292
// MI455X (gfx1250) — compile-verified
//
#include <hip/hip_runtime.h>

typedef float v2f __attribute__((ext_vector_type(2)));
typedef float v8f __attribute__((ext_vector_type(8)));

#define NQ   8
#define DIM  256      // 2^8
#define NW   512      // re/im interleaved output columns
#define MT   32       // rows per workgroup (2 WMMA M-tiles)

// ---------------------------------------------------------------------------
// Kernel A: build the circuit matrix W in workspace, pre-swizzled for WMMA.
//
// Logical W[k][n], k=0..255 (input basis), n=2m+ri (output amplitude m, ri=0 re / 1 im):
//   W[k][2m+0] = Re( U[m,k] * (-i)^popc(k) )
//   W[k][2m+1] = Im( U[m,k] * (-i)^popc(k) )
// Stored as float2 pairs over K:  ws[k2*NW + n] = { W[2k2][n], W[2k2+1][n] }
// so the 16x16x4 B-fragment (k = k0+2*half, k0+2*half+1) is ONE b64 load.
// ---------------------------------------------------------------------------
__global__ __launch_bounds__(128) void build_unitary(
    const float* __restrict__ weights,   // (6,8,3)
    float* __restrict__ W)               // 256*512 floats (512 KB)
{
    __shared__ float sRe[DIM];
    __shared__ float sIm[DIM];
    const int k = blockIdx.x;
    const int t = threadIdx.x;           // 0..127

    for (int i = t; i < DIM; i += 128) { sRe[i] = (i == k) ? 1.f : 0.f; sIm[i] = 0.f; }

    for (int l = 0; l < 6; ++l) {
        // ---- Rot(phi,theta,omega) on each wire (wire q = bit 7-q, MSB first) ----
        for (int q = 0; q < NQ; ++q) {
            const float phi   = weights[(l*NQ + q)*3 + 0];
            const float theta = weights[(l*NQ + q)*3 + 1];
            const float omega = weights[(l*NQ + q)*3 + 2];
            float st, ct;  __sincosf(0.5f*theta, &st, &ct);
            float sA, cA;  __sincosf(0.5f*(phi+omega), &sA, &cA);
            float sB, cB;  __sincosf(0.5f*(phi-omega), &sB, &cB);
            const float u00r =  cA*ct, u00i = -sA*ct;
            const float u01r = -cB*st, u01i = -sB*st;
            const float u10r =  cB*st, u10i = -sB*st;
            const float u11r =  cA*ct, u11i =  sA*ct;

            const int p = 7 - q;
            const unsigned mask = 1u << p;
            const unsigned i0 = ((t & ~(mask - 1u)) << 1) | (t & (mask - 1u));
            const unsigned i1 = i0 | mask;
            __syncthreads();
            const float a0r = sRe[i0], a0i = sIm[i0];
            const float a1r = sRe[i1], a1i = sIm[i1];
            sRe[i0] = u00r*a0r - u00i*a0i + u01r*a1r - u01i*a1i;
            sIm[i0] = u00r*a0i + u00i*a0r + u01r*a1i + u01i*a1r;
            sRe[i1] = u10r*a0r - u10i*a0i + u11r*a1r - u11i*a1i;
            sIm[i1] = u10r*a0i + u10i*a0r + u11r*a1i + u11i*a1r;
        }
        // ---- CNOT ring with range r = l%7 + 1 ----
        const int rr = (l % 7) + 1;
        for (int q = 0; q < NQ; ++q) {
            const int cb = 7 - q;
            const int tb = 7 - ((q + rr) & 7);
            __syncthreads();
            if (t < 64) {
                const int hi = cb > tb ? cb : tb;
                const int lo = cb > tb ? tb : cb;
                const unsigned lowb  = (unsigned)t & ((1u << lo) - 1u);
                const unsigned midb  = ((unsigned)t >> lo) & ((1u << (hi - lo - 1)) - 1u);
                const unsigned highb = (unsigned)t >> (hi - 1);
                unsigned i = (highb << (hi + 1)) | (midb << (lo + 1)) | lowb | (1u << cb);
                const unsigned j = i | (1u << tb);   // i has target bit 0
                const float tr = sRe[i], ti = sIm[i];
                sRe[i] = sRe[j]; sIm[i] = sIm[j];
                sRe[j] = tr;     sIm[j] = ti;
            }
        }
    }

    // ---- multiply column by (-i)^popc(k) and store swizzled ----
    __syncthreads();
    const int pc = __popc(k) & 3;
    const int k2 = k >> 1, par = k & 1;
    float* base = W + (size_t)k2 * (2 * NW);
    for (int m = t; m < DIM; m += 128) {
        const float re = sRe[m], im = sIm[m];
        float wr, wi;
        if      (pc == 0) { wr =  re; wi =  im; }
        else if (pc == 1) { wr =  im; wi = -re; }   // *( -i )
        else if (pc == 2) { wr = -re; wi = -im; }
        else              { wr = -im; wi =  re; }
        base[(2*m + 0)*2 + par] = wr;   // n = 2m     (Re column)
        base[(2*m + 1)*2 + par] = wi;   // n = 2m + 1 (Im column)
    }
}

// ---------------------------------------------------------------------------
// Kernel B: per 32-row tile, build the real amplitude matrix v[r][k] in LDS,
// GEMM (32 x 256) x (256 x 512) with v_wmma_f32_16x16x4_f32 (each wave owns
// 2 M-tiles x 4 N-tiles; B fragments shared across M-tiles), then reduce
// |psi|^2 -> 8 Z expectations per row.
//
// LDS overlay: the A-tile (2 x [256][16] = 8192 floats) and the probability
// array ([32][260] = 8320 floats) are live in disjoint phases separated by a
// barrier, so they share one buffer.
// ---------------------------------------------------------------------------
__global__ __launch_bounds__(256) void qembed_gemm_z(
    const float* __restrict__ x,     // (B, 8)
    const float* __restrict__ W,     // swizzled, see above
    float* __restrict__ out)         // (B, 8)
{
    __shared__ float smem[MT * (DIM + 4)];   // 8320 floats: A-tile then probs
    __shared__ float csc[MT][NQ];
    __shared__ float css[MT][NQ];

    float* const As    = smem;               // As[mt*4096 + k*16 + r], r=0..15
    float* const probs = smem;               // probs[r*(DIM+4) + m],   r=0..31

    const int tid  = threadIdx.x;
    const int row0 = blockIdx.x * MT;

    // --- per-row sin/cos: 256 threads = 32 rows x 8 qubits ---
    {
        const int r = tid >> 3, q = tid & 7;
        float s, c;
        __sincosf(0.5f * x[(size_t)(row0 + r) * NQ + q], &s, &c);
        csc[r][q] = c;  css[r][q] = s;
    }
    __syncthreads();

    // --- A tile: v[r][k] = prod_q (bit_{7-q}(k) ? sin : cos) ---
    for (int idx = tid; idx < MT * DIM; idx += 256) {
        const int r5 = idx & 31;             // global row in tile
        const int k  = idx >> 5;
        float p = 1.f;
#pragma unroll
        for (int q = 0; q < NQ; ++q)
            p *= ((k >> (7 - q)) & 1) ? css[r5][q] : csc[r5][q];
        As[(r5 >> 4) * (DIM * 16) + k * 16 + (r5 & 15)] = p;
    }
    __syncthreads();

    const int lane = tid & 31;
    const int wave = tid >> 5;      // 0..7, each owns 4 N-tiles of 16
    const int col  = lane & 15;
    const int hf   = lane >> 4;

    const v8f vz = {0.f,0.f,0.f,0.f,0.f,0.f,0.f,0.f};
    v8f a00 = vz, a01 = vz, a02 = vz, a03 = vz;   // M-tile 0, N-tiles 0..3
    v8f a10 = vz, a11 = vz, a12 = vz, a13 = vz;   // M-tile 1, N-tiles 0..3
    const int n0 = wave * 64 + col;               // base output column for this lane
    const float2* __restrict__ Wp = (const float2*)W;

#pragma unroll 4
    for (int k0 = 0; k0 < DIM; k0 += 4) {
        const int ka = k0 + 2 * hf;
        v2f fa0, fa1;                             // A frags for the two M-tiles
        fa0.x = As[ka * 16 + col];
        fa0.y = As[(ka + 1) * 16 + col];
        fa1.x = As[DIM * 16 + ka * 16 + col];
        fa1.y = As[DIM * 16 + (ka + 1) * 16 + col];
        const float2* wrow = Wp + (size_t)((k0 >> 1) + hf) * NW;
        const float2 w0 = wrow[n0 +  0];
        const float2 w1 = wrow[n0 + 16];
        const float2 w2 = wrow[n0 + 32];
        const float2 w3 = wrow[n0 + 48];
        v2f b0; b0.x = w0.x; b0.y = w0.y;
        v2f b1; b1.x = w1.x; b1.y = w1.y;
        v2f b2; b2.x = w2.x; b2.y = w2.y;
        v2f b3; b3.x = w3.x; b3.y = w3.y;
        a00 = __builtin_amdgcn_wmma_f32_16x16x4_f32(false, fa0, false, b0, (short)0, a00, false, false);
        a10 = __builtin_amdgcn_wmma_f32_16x16x4_f32(false, fa1, false, b0, (short)0, a10, false, false);
        a01 = __builtin_amdgcn_wmma_f32_16x16x4_f32(false, fa0, false, b1, (short)0, a01, false, false);
        a11 = __builtin_amdgcn_wmma_f32_16x16x4_f32(false, fa1, false, b1, (short)0, a11, false, false);
        a02 = __builtin_amdgcn_wmma_f32_16x16x4_f32(false, fa0, false, b2, (short)0, a02, false, false);
        a12 = __builtin_amdgcn_wmma_f32_16x16x4_f32(false, fa1, false, b2, (short)0, a12, false, false);
        a03 = __builtin_amdgcn_wmma_f32_16x16x4_f32(false, fa0, false, b3, (short)0, a03, false, false);
        a13 = __builtin_amdgcn_wmma_f32_16x16x4_f32(false, fa1, false, b3, (short)0, a13, false, false);
    }

    // A-tile no longer needed; probs overlays the same LDS buffer.
    __syncthreads();

    // --- |psi|^2: square, sum re/im lane pairs, scatter into probs ---
    v8f accs[2][4] = {{a00, a01, a02, a03}, {a10, a11, a12, a13}};
#pragma unroll
    for (int mt = 0; mt < 2; ++mt) {
#pragma unroll
        for (int j = 0; j < 4; ++j) {
            const int nt = wave * 4 + j;          // N-tile; columns nt*16 .. nt*16+15
#pragma unroll
            for (int v = 0; v < 8; ++v) {
                const float cv = accs[mt][j][v];
                const float sq = cv * cv;
                const float pm = sq + __shfl_xor(sq, 1, 32);   // re^2 + im^2
                if ((col & 1) == 0)
                    probs[(mt * 16 + v + 8 * hf) * (DIM + 4) + nt * 8 + (col >> 1)] = pm;
            }
        }
    }
    __syncthreads();

    // --- Z_i = sum_m p_m * (1 - 2*bit_{7-i}(m)); 256 threads = 32 rows x 8 qubits ---
    {
        const int r = tid >> 3;         // row in tile
        const int i = tid & 7;          // qubit
        const int bp = 7 - i;
        float z = 0.f;
        for (int m = 0; m < DIM; ++m) {
            const float pm = probs[r * (DIM + 4) + m];
            z += ((m >> bp) & 1) ? -pm : pm;
        }
        out[(size_t)(row0 + r) * NQ + i] = z;
    }
}

// ---------------------------------------------------------------------------
extern "C" void kernel_launch(void* const* d_in, const int* in_sizes, int n_in,
                              void* d_out, int out_size, void* d_ws, size_t ws_size,
                              hipStream_t stream) {
    const float* inputs  = (const float*)d_in[0];   // (B, 8) float32
    const float* weights = (const float*)d_in[1];   // (6, 8, 3) float32
    float* out = (float*)d_out;
    float* W   = (float*)d_ws;                      // needs 256*512*4 = 512 KB

    const int B = in_sizes[0] / NQ;                 // 65536

    build_unitary<<<DIM, 128, 0, stream>>>(weights, W);
    qembed_gemm_z<<<B / MT, 256, 0, stream>>>(inputs, W, out);
}